// RFNetwork_27023934226791
// MI455X (gfx1250) — compile-verified
//
#include <hip/hip_runtime.h>
#include <math.h>

#define T_STEPS   256
#define NIN       4096
#define NOUT      2048
#define KIN       204      // 4*51
#define KOUT      102
#define LMBDA     0.01f
#define MAXPOST   5.0f
#define MAXPRE    5.0f

#define NB        64       // persistent blocks (co-resident)
#define NT        512      // 16 wave32 waves per block
#define WPB       (NT / 32)

typedef float v2f __attribute__((ext_vector_type(2)));
typedef float v8f __attribute__((ext_vector_type(8)));

// ---------------- deterministic RNG (stand-in for jax threefry normal) ----
__device__ __forceinline__ unsigned hash_u32(unsigned x) {
  x ^= x >> 16; x *= 0x7feb352dU;
  x ^= x >> 15; x *= 0x846ca68bU;
  x ^= x >> 16; return x;
}
__device__ __forceinline__ float rnorm(unsigned sid, unsigned i) {
  unsigned h1 = hash_u32(sid * 0x9E3779B9u ^ hash_u32(i + 0x85EBCA6Bu));
  unsigned h2 = hash_u32(h1 + 0xC2B2AE35u);
  float u1 = (float)((h1 >> 8) + 1u) * (1.0f / 16777216.0f);   // (0,1]
  float u2 = (float)(h2 >> 8)        * (1.0f / 16777216.0f);   // [0,1)
  return sqrtf(-2.0f * __logf(u1)) * __cosf(6.28318530717958647f * u2);
}

// sortable key: larger float -> larger unsigned
__device__ __forceinline__ unsigned fkey(float f) {
  unsigned u = __float_as_uint(f);
  return (u & 0x80000000u) ? ~u : (u | 0x80000000u);
}

// ---------------- device-wide barrier (arrive + generation release) -------
__device__ __forceinline__ void grid_barrier(unsigned* cnt, volatile unsigned* gen) {
  __syncthreads();
  __threadfence();
  if (threadIdx.x == 0) {
    unsigned g = *gen;
    if (atomicAdd(cnt, 1u) == (unsigned)(NB - 1)) {
      *cnt = 0u;
      __threadfence();
      *gen = g + 1u;                       // release
    } else {
      while (*gen == g) { __builtin_amdgcn_s_sleep(1); }
    }
  }
  __syncthreads();
  __threadfence();
}

// ---------------- block-wide top-k radix select (block 0 only) ------------
__device__ void topk_block(const float* vals, int n, int k,
                           int* __restrict__ idx_out, float* __restrict__ bin_out,
                           unsigned* hist, unsigned* sg, unsigned* se, unsigned* bc) {
  const int tid = threadIdx.x;
  const int epb = n / NT;                  // 8 (input) or 4 (output)
  unsigned mask = 0u, prefix = 0u, krem = (unsigned)k, cntgt = 0u;

  for (int round = 3; round >= 0; --round) {
    const int shift = round * 8;
    for (int b = tid; b < 256; b += NT) hist[b] = 0u;
    __syncthreads();
    for (int e = 0; e < epb; ++e) {
      unsigned key = fkey(vals[tid * epb + e]);
      if ((key & mask) == prefix) atomicAdd(&hist[(key >> shift) & 255], 1u);
    }
    __syncthreads();
    if (tid == 0) {
      unsigned cum = 0u; int b = 255;
      for (; b >= 0; --b) {
        unsigned h = hist[b];
        if (cum + h >= krem) break;
        cum += h;
      }
      bc[0] = prefix | ((unsigned)b << shift);
      bc[1] = krem - cum;
      bc[2] = cntgt + cum;
    }
    __syncthreads();
    prefix = bc[0]; krem = bc[1]; cntgt = bc[2];
    mask |= (0xFFu << shift);
    __syncthreads();
  }
  const unsigned Tkey    = prefix;
  const unsigned need_eq = krem;    // winners taken from == Tkey, index order
  const unsigned base_eq = cntgt;   // number of strictly-greater winners

  unsigned cg = 0u, ce = 0u;
  for (int e = 0; e < epb; ++e) {
    unsigned key = fkey(vals[tid * epb + e]);
    cg += (key > Tkey); ce += (key == Tkey);
  }
  sg[tid] = cg; se[tid] = ce;
  __syncthreads();
  if (tid == 0) {
    unsigned ag = 0u, ae = 0u;
    for (int q = 0; q < NT; ++q) {
      unsigned tg = sg[q], te = se[q];
      sg[q] = ag; se[q] = ae;
      ag += tg; ae += te;
    }
  }
  __syncthreads();
  unsigned gpos = sg[tid], epos = se[tid];
  for (int e = 0; e < epb; ++e) {
    int i = tid * epb + e;
    unsigned key = fkey(vals[i]);
    bool sel = false;
    if (key > Tkey) { idx_out[gpos++] = i; sel = true; }
    else if (key == Tkey) {
      if (epos < need_eq) { idx_out[base_eq + epos] = i; sel = true; }
      epos++;
    }
    if (bin_out) bin_out[i] = sel ? 1.0f : 0.0f;
  }
  __syncthreads();
}

// ---------------- persistent kernel: whole 256-step recurrence ------------
__global__ __launch_bounds__(NT, 1) void rf_persistent(
    const float* __restrict__ input,    // [T, NIN]
    const float* __restrict__ fixedW,   // [NOUT, NIN]
    float* __restrict__ out,            // [T, NOUT]
    float* __restrict__ plastic,        // [NOUT, NIN] (zeroed)
    float* __restrict__ out_hat,        // [NOUT]
    float* __restrict__ rowscale,       // [NOUT]
    float* __restrict__ colscale,       // [NIN]
    float* __restrict__ rowsum,         // [NOUT] (zeroed: sums of P=0)
    float* __restrict__ partials,       // [NB, NIN] column partial sums
    int*   __restrict__ in_idx,         // [KIN]
    int*   __restrict__ out_idx,        // [KOUT]
    unsigned* bar_cnt, unsigned* bar_gen) {

  __shared__ union {
    struct {
      float    xn[NIN];        // noisy copy for selection (16 KB)
      unsigned hist[256];
      unsigned sg[NT];
      unsigned se[NT];
      float    redA[NT];
      float    redB[NT];
    } sel;
    float cs[8][64];           // column partial reduce scratch
  } sh;
  __shared__ unsigned bc[4];

  const int tid  = threadIdx.x;
  const int lane = tid & 31;

#pragma unroll 1
  for (int t = 0; t < T_STEPS; ++t) {
    // ---- phase 1: input WTA (block 0) -----------------------------------
    if (blockIdx.x == 0) {
      const float* x = input + (size_t)t * NIN;
      float mx = -3.4e38f, mn = 3.4e38f;
      for (int i = tid; i < NIN; i += NT) {
        float v = x[i]; mx = fmaxf(mx, v); mn = fminf(mn, v);
      }
      sh.sel.redA[tid] = mx; sh.sel.redB[tid] = mn;
      __syncthreads();
      for (int o = NT / 2; o; o >>= 1) {
        if (tid < o) {
          sh.sel.redA[tid] = fmaxf(sh.sel.redA[tid], sh.sel.redA[tid + o]);
          sh.sel.redB[tid] = fminf(sh.sel.redB[tid], sh.sel.redB[tid + o]);
        }
        __syncthreads();
      }
      float scale = (1e-10f + sh.sel.redA[0] - sh.sel.redB[0]) * 0.1f;
      for (int i = tid; i < NIN; i += NT)
        sh.sel.xn[i] = x[i] + scale * rnorm(2u * (unsigned)t, (unsigned)i);
      __syncthreads();
      topk_block(sh.sel.xn, NIN, KIN, in_idx, nullptr,
                 sh.sel.hist, sh.sel.sg, sh.sel.se, bc);
    }
    grid_barrier(bar_cnt, bar_gen);

    // ---- phase 2: out_hat = (fixed + plastic) @ in_  via WMMA f32 -------
    {
      int gw = blockIdx.x * WPB + (tid >> 5);
      if (gw < NOUT / 16) {
        const int half = lane >> 4;        // lanes 0-15: K0/K1, lanes 16-31: K2/K3
        const int m    = lane & 15;
        const int row  = gw * 16 + m;
        const float* fr = fixedW  + (size_t)row * NIN;
        const float* pr = plastic + (size_t)row * NIN;
        v2f bmat;                          // B[k][n] = (n==0); n = lane&15
        bmat.x = (m == 0) ? 1.0f : 0.0f;
        bmat.y = bmat.x;
        v8f acc = {0.f, 0.f, 0.f, 0.f, 0.f, 0.f, 0.f, 0.f};
        for (int kk = 0; kk < KIN / 4; ++kk) {
          int c0 = in_idx[kk * 4 + half * 2 + 0];
          int c1 = in_idx[kk * 4 + half * 2 + 1];
          v2f amat;
          amat.x = fr[c0] + pr[c0];
          amat.y = fr[c1] + pr[c1];
          acc = __builtin_amdgcn_wmma_f32_16x16x4_f32(
              false, amat, false, bmat, (short)0, acc, false, false);
        }
        if (m == 0) {                      // C column N=0: lane0 M=0..7, lane16 M=8..15
          int base = gw * 16 + half * 8;
#pragma unroll
          for (int r = 0; r < 8; ++r) out_hat[base + r] = acc[r];
        }
      }
    }
    grid_barrier(bar_cnt, bar_gen);

    // ---- phase 3: output WTA (block 0), writes binary row of d_out ------
    if (blockIdx.x == 0) {
      float mn = 3.4e38f;
      for (int i = tid; i < NOUT; i += NT) mn = fminf(mn, out_hat[i]);
      sh.sel.redB[tid] = mn;
      __syncthreads();
      for (int o = NT / 2; o; o >>= 1) {
        if (tid < o) sh.sel.redB[tid] = fminf(sh.sel.redB[tid], sh.sel.redB[tid + o]);
        __syncthreads();
      }
      float scale = fabsf(sh.sel.redB[0] * 0.1f);
      for (int i = tid; i < NOUT; i += NT)
        sh.sel.xn[i] = out_hat[i] + scale * rnorm(2u * (unsigned)t + 1u, (unsigned)i);
      __syncthreads();
      topk_block(sh.sel.xn, NOUT, KOUT, out_idx, out + (size_t)t * NOUT,
                 sh.sel.hist, sh.sel.sg, sh.sel.se, bc);
    }
    grid_barrier(bar_cnt, bar_gen);

    // ---- phase 4: Hebbian update (unique writers) + rowscale ------------
    // Row sums after the rank-1 update are analytic: selected rows gain
    // exactly LMBDA*KIN; rowsum[] holds sums of the renormed matrix from
    // the previous step (zeros at t=0).
    {
      int gidx = blockIdx.x * NT + tid;               // 32768 threads
      if (gidx < KOUT * KIN) {
        int oi = out_idx[gidx / KIN];
        int ij = in_idx[gidx % KIN];
        plastic[(size_t)oi * NIN + ij] += LMBDA;
      }
      if (gidx < NOUT) {
        float s = rowsum[gidx] + out[(size_t)t * NOUT + gidx] * (LMBDA * (float)KIN);
        rowscale[gidx] = (s > MAXPOST) ? (MAXPOST / s) : 1.0f;
      }
    }
    grid_barrier(bar_cnt, bar_gen);

    // ---- phase 5: FUSED row-apply + column partial sums -----------------
    // Block owns a 32-row stripe; thread owns 8 unique columns (tid+k*512),
    // so column partials stay in registers (no atomics, deterministic).
    {
      const int row0 = blockIdx.x * (NOUT / NB);      // 32 rows per block
      float acc[8];
#pragma unroll
      for (int k2 = 0; k2 < 8; ++k2) acc[k2] = 0.f;
      for (int r = 0; r < NOUT / NB; ++r) {
        const int row = row0 + r;
        const float rs = rowscale[row];
        const size_t base = (size_t)row * NIN;
#pragma unroll
        for (int k2 = 0; k2 < 8; ++k2) {
          const int c = tid + k2 * NT;
          float v = plastic[base + c] * rs;
          plastic[base + c] = v;
          acc[k2] += v;
        }
      }
#pragma unroll
      for (int k2 = 0; k2 < 8; ++k2)
        partials[(size_t)blockIdx.x * NIN + tid + k2 * NT] = acc[k2];
    }
    grid_barrier(bar_cnt, bar_gen);

    // ---- phase 6: reduce column partials -> colscale (1 MB pass) --------
    {
      const int tx = tid & 63, ty = tid >> 6;         // 64 cols x 8 strips
      const int col = blockIdx.x * 64 + tx;
      float s = 0.f;
      for (int p = ty; p < NB; p += 8) s += partials[(size_t)p * NIN + col];
      sh.cs[ty][tx] = s;
      __syncthreads();
      if (ty == 0) {
        float tot = 0.f;
#pragma unroll
        for (int q = 0; q < 8; ++q) tot += sh.cs[q][tx];
        colscale[col] = (tot > MAXPRE) ? (MAXPRE / tot) : 1.0f;
      }
      __syncthreads();
    }
    grid_barrier(bar_cnt, bar_gen);

    // ---- phase 7: FUSED col-apply + row sums for next step --------------
    // Wave owns 2 full rows: apply colscale, store, and shfl-reduce the
    // renormed row sum (feeds phase 4 of step t+1).
    {
      const int gwave = blockIdx.x * WPB + (tid >> 5);  // 0..1023
#pragma unroll
      for (int rr = 0; rr < 2; ++rr) {
        const int row = gwave * 2 + rr;
        const size_t base = (size_t)row * NIN;
        float s = 0.f;
        for (int c = lane; c < NIN; c += 32) {
          float v = plastic[base + c] * colscale[c];
          plastic[base + c] = v;
          s += v;
        }
        for (int o = 16; o; o >>= 1) s += __shfl_xor(s, o, 32);
        if (lane == 0) rowsum[row] = s;
      }
    }
    grid_barrier(bar_cnt, bar_gen);
  }
}

// ---------------- host-side launch ----------------------------------------
extern "C" void kernel_launch(void* const* d_in, const int* in_sizes, int n_in,
                              void* d_out, int out_size, void* d_ws, size_t ws_size,
                              hipStream_t stream) {
  (void)in_sizes; (void)n_in; (void)out_size; (void)ws_size;
  const float* input  = (const float*)d_in[0];   // [256, 4096] f32
  const float* fixedW = (const float*)d_in[1];   // [2048, 4096] f32
  float* out = (float*)d_out;                    // [256, 2048] f32

  char* ws = (char*)d_ws;
  size_t off = 0;
  float* plastic  = (float*)(ws + off); off += (size_t)NOUT * NIN * sizeof(float);
  float* out_hat  = (float*)(ws + off); off += NOUT * sizeof(float);
  float* rowscale = (float*)(ws + off); off += NOUT * sizeof(float);
  float* colscale = (float*)(ws + off); off += NIN * sizeof(float);
  float* rowsum   = (float*)(ws + off); off += NOUT * sizeof(float);
  float* partials = (float*)(ws + off); off += (size_t)NB * NIN * sizeof(float);
  int* in_idx     = (int*)(ws + off);   off += 256 * sizeof(int);
  int* out_idx    = (int*)(ws + off);   off += 128 * sizeof(int);
  off = (off + 127) & ~(size_t)127;
  unsigned* bar_cnt = (unsigned*)(ws + off); off += 128;   // own cache line
  unsigned* bar_gen = (unsigned*)(ws + off); off += 128;   // own cache line

  // Zero plastic + rowsum + barrier state (deterministic start each call).
  hipMemsetAsync(d_ws, 0, off, stream);

  rf_persistent<<<NB, NT, 0, stream>>>(input, fixedW, out, plastic, out_hat,
                                       rowscale, colscale, rowsum, partials,
                                       in_idx, out_idx, bar_cnt, bar_gen);
}